// HessBlock_48533130445362
// MI455X (gfx1250) — compile-verified
//
#include <hip/hip_runtime.h>
#include <hip/hip_bf16.h>

typedef __attribute__((ext_vector_type(2))) float v2f;
typedef __attribute__((ext_vector_type(8))) float v8f;

#define NSIDE 128
#define NMASK 127
#define KS 13
#define KH 6

// ds_swizzle XOR mode within 32 lanes: and_mask=0x1f, or_mask=0, xor_mask=m
#define SWZ_XOR(v, m) \
    __int_as_float(__builtin_amdgcn_ds_swizzle(__float_as_int(v), 0x7c00 | (m)))

__device__ __forceinline__ float ld3(const float* __restrict__ s, int b, int d, int h, int w) {
    return s[(((b * NSIDE + d) * NSIDE + h) * NSIDE) + w];
}

// value of s at point with axis `ax` coordinate replaced by c (0=D,1=H,2=W)
__device__ __forceinline__ float at_ax(const float* __restrict__ s, int b, int d, int h, int w,
                                       int ax, int c) {
    if (ax == 0) d = c; else if (ax == 1) h = c; else w = c;
    return ld3(s, b, d, h, w);
}

// torch.gradient along `ax`, branchless: (f[min(c+1)]-f[max(c-1)]) * (interior ? 0.5 : 1)
__device__ __forceinline__ float tg1(const float* __restrict__ s, int b, int d, int h, int w,
                                     int ax) {
    int c = (ax == 0) ? d : (ax == 1) ? h : w;
    int cp = c + 1 > NMASK ? NMASK : c + 1;
    int cm = c - 1 < 0 ? 0 : c - 1;
    float sc = (cp - cm == 2) ? 0.5f : 1.0f;
    return (at_ax(s, b, d, h, w, ax, cp) - at_ax(s, b, d, h, w, ax, cm)) * sc;
}

// tg1 along axA at a point whose axB coordinate is replaced by cB
__device__ __forceinline__ float tg1_at(const float* __restrict__ s, int b, int d, int h, int w,
                                        int axA, int axB, int cB) {
    if (axB == 0) d = cB; else if (axB == 1) h = cB; else w = cB;
    return tg1(s, b, d, h, w, axA);
}

// Hessian element: outer tgrad along axB of inner tgrad along axA (exact edge composition)
__device__ __forceinline__ float hess(const float* __restrict__ s, int b, int d, int h, int w,
                                      int axA, int axB) {
    int c = (axB == 0) ? d : (axB == 1) ? h : w;
    int cp = c + 1 > NMASK ? NMASK : c + 1;
    int cm = c - 1 < 0 ? 0 : c - 1;
    float sc = (cp - cm == 2) ? 0.5f : 1.0f;
    return (tg1_at(s, b, d, h, w, axA, axB, cp) - tg1_at(s, b, d, h, w, axA, axB, cm)) * sc;
}

// One-time: normalized 13-tap Gaussian weights (sigma = clamp(scale, 0.1, 10))
__global__ void gauss_weights_kernel(const float* __restrict__ scale_p,
                                     float* __restrict__ wt) {
    if (threadIdx.x == 0) {
        float sigma = fminf(fmaxf(scale_p[0], 0.1f), 10.0f);
        float inv2s2 = 1.0f / (2.0f * sigma * sigma);
        float e[KS];
        float sum = 0.0f;
#pragma unroll
        for (int t = 0; t < KS; ++t) {
            float x = (float)(t - KH);
            e[t] = __expf(-x * x * inv2s2);
            sum += e[t];
        }
        float rs = 1.0f / sum;
#pragma unroll
        for (int t = 0; t < KS; ++t) wt[t] = e[t] * rs;
    }
}

// 13-tap separable blur along AXIS with replicate (clamp) padding; weights via scalar loads.
template <int AXIS>
__global__ void __launch_bounds__(256) blur1d_kernel(const float* __restrict__ in,
                                                     float* __restrict__ out,
                                                     const float* __restrict__ wtab) {
    int idx = blockIdx.x * blockDim.x + threadIdx.x;
    int w = idx & NMASK;
    int h = (idx >> 7) & NMASK;
    int d = (idx >> 14) & NMASK;
    int b = idx >> 21;

    float wt[KS];
#pragma unroll
    for (int t = 0; t < KS; ++t) wt[t] = wtab[t];   // uniform -> SMEM loads

    int c0 = (AXIS == 0) ? d : (AXIS == 1) ? h : w;
    float acc = 0.0f;
#pragma unroll
    for (int t = 0; t < KS; ++t) {
        int c = c0 + t - KH;
        c = c < 0 ? 0 : (c > NMASK ? NMASK : c);
        acc += wt[t] * at_ax(in, b, d, h, w, AXIS, c);
    }
    out[idx] = acc;
}

// Fused Hessian features + MLP. One wave = 16 voxels.
// Layer 1 (7->10) = two f32 K=4 WMMAs; layer 2 (10->1) = merging ds_swizzle butterfly.
__global__ void __launch_bounds__(256) hess_mlp_kernel(
        const float* __restrict__ smooth, const float* __restrict__ scale_p,
        const float* __restrict__ W1, const float* __restrict__ b1p,
        const float* __restrict__ W2, const float* __restrict__ b2p,
        float* __restrict__ out) {
    const int lane = threadIdx.x & 31;
    const int half = lane >> 4;       // 0: K={0,1}/{4,5}, 1: K={2,3}/{6,7}
    const int n = lane & 15;          // A row (voxel-in-tile) AND B/C column
    const int waveInBlock = threadIdx.x >> 5;
    const int voxbase = (blockIdx.x * (blockDim.x >> 5) + waveInBlock) * 16;

    const int vox = voxbase + n;
    const int w = vox & NMASK;
    const int h = (vox >> 7) & NMASK;
    const int d = (vox >> 14) & NMASK;
    const int b = vox >> 21;

    const float scl = scale_p[0];     // raw scale is the 7th feature

    // f0=h_DD f1=h_DH f2=h_DW f3=h_HH f4=h_HW f5=h_WW f6=scale f7=0
    float a1x, a1y, a2x, a2y;
    if (half == 0) {
        a1x = hess(smooth, b, d, h, w, 0, 0);   // K=0
        a1y = hess(smooth, b, d, h, w, 0, 1);   // K=1
        a2x = hess(smooth, b, d, h, w, 1, 2);   // K=4
        a2y = hess(smooth, b, d, h, w, 2, 2);   // K=5
    } else {
        a1x = hess(smooth, b, d, h, w, 0, 2);   // K=2
        a1y = hess(smooth, b, d, h, w, 1, 1);   // K=3
        a2x = scl;                              // K=6
        a2y = 0.0f;                             // K=7 (pad)
    }

    // B = W1^T padded to 4x16 chunks: VGPR v, lanes0-15 -> K=v, lanes16-31 -> K=v+2.
    // Branchless gathers: clamp row index, then cndmask to zero for n >= 10.
    const int k1a = half ? 2 : 0, k1b = half ? 3 : 1;
    const int k2a = half ? 6 : 4, k2b = half ? 7 : 5;
    const bool colv = (n < 10);
    const int nc = colv ? n : 9;
    float b1x = W1[nc * 7 + k1a];
    float b1y = W1[nc * 7 + k1b];
    float b2x = W1[nc * 7 + k2a];
    float b2y = (k2b < 7) ? W1[nc * 7 + 5] : 0.0f;   // k2b is 5 or 7
    float bias = b1p[nc];
    float w2n  = W2[nc];
    if (!colv) { b1x = 0.0f; b1y = 0.0f; b2x = 0.0f; b2y = 0.0f; bias = 0.0f; w2n = 0.0f; }

    v8f c;
#pragma unroll
    for (int v = 0; v < 8; ++v) c[v] = bias;

    v2f A1; A1.x = a1x; A1.y = a1y;
    v2f B1; B1.x = b1x; B1.y = b1y;
    v2f A2; A2.x = a2x; A2.y = a2y;
    v2f B2; B2.x = b2x; B2.y = b2y;

    c = __builtin_amdgcn_wmma_f32_16x16x4_f32(false, A1, false, B1, (short)0, c, false, false);
    c = __builtin_amdgcn_wmma_f32_16x16x4_f32(false, A2, false, B2, (short)0, c, false, false);

    // Layer 2 partials: lane holds column n of rows v (+8*half).
    float p[8];
#pragma unroll
    for (int v = 0; v < 8; ++v) p[v] = fmaxf(c[v], 0.0f) * w2n;

    // Merging butterfly over the 16-lane half: 4+2+1+1 = 8 swizzles.
    // After each xor-k step, live values halve and row index absorbs bit k.
    const int b0 = n & 1, b1b = (n >> 1) & 1, b2b = (n >> 2) & 1;
    float q[4];
#pragma unroll
    for (int u = 0; u < 4; ++u) {
        float send = b0 ? p[2 * u] : p[2 * u + 1];
        float keep = b0 ? p[2 * u + 1] : p[2 * u];
        q[u] = keep + SWZ_XOR(send, 1);          // row 2u + b0, pair-summed
    }
    float r[2];
#pragma unroll
    for (int u = 0; u < 2; ++u) {
        float send = b1b ? q[2 * u] : q[2 * u + 1];
        float keep = b1b ? q[2 * u + 1] : q[2 * u];
        r[u] = keep + SWZ_XOR(send, 2);          // row 4u + 2*b1 + b0
    }
    {
        float send = b2b ? r[0] : r[1];
        float keep = b2b ? r[1] : r[0];
        float s = keep + SWZ_XOR(send, 4);       // row (n & 7), 8-lane summed
        s += SWZ_XOR(s, 8);                      // full 16-lane sum of this half
        if (n < 8) {                             // lane j holds row j of its half
            const float b2v = b2p[0];
            out[voxbase + n + 8 * half] = 1.0f / (1.0f + __expf(-(s + b2v)));
        }
    }
}

extern "C" void kernel_launch(void* const* d_in, const int* in_sizes, int n_in,
                              void* d_out, int out_size, void* d_ws, size_t ws_size,
                              hipStream_t stream) {
    const float* x     = (const float*)d_in[0];
    const float* scale = (const float*)d_in[1];
    const float* W1    = (const float*)d_in[2];
    const float* b1    = (const float*)d_in[3];
    const float* W2    = (const float*)d_in[4];
    const float* b2    = (const float*)d_in[5];
    float* out = (float*)d_out;

    const int V = 2 * NSIDE * NSIDE * NSIDE;     // 4,194,304 voxels
    float* buf0 = (float*)d_ws;
    float* buf1 = buf0 + V;
    float* wtab = buf1 + V;                      // 13 floats of normalized taps

    gauss_weights_kernel<<<1, 32, 0, stream>>>(scale, wtab);

    dim3 blk(256);
    dim3 grdB(V / 256);
    // Separable Gaussian blur (clamp padding per axis == 3D replicate-pad conv exactly)
    blur1d_kernel<2><<<grdB, blk, 0, stream>>>(x,    buf0, wtab);  // along W
    blur1d_kernel<1><<<grdB, blk, 0, stream>>>(buf0, buf1, wtab);  // along H
    blur1d_kernel<0><<<grdB, blk, 0, stream>>>(buf1, buf0, wtab);  // along D -> smooth

    // Fused Hessian + MLP: 16 voxels/wave, 8 waves/block -> 128 voxels/block
    hess_mlp_kernel<<<V / 128, 256, 0, stream>>>(buf0, scale, W1, b1, W2, b2, out);
}